// Net_89945205113615
// MI455X (gfx1250) — compile-verified
//
#include <hip/hip_runtime.h>
#include <math.h>

// ---------------------------------------------------------------------------
// 3-layer GCN inference for MI455X (gfx1250).
// Memory-bound workload: edge gather/scatter dominates; h/agg arrays
// (12.8-25.6MB) are L2-resident (192MB L2). GEMMs (~1 GFLOP total) run on the
// matrix pipe with full-precision V_WMMA_F32_16X16X4_F32.
// All shapes are compile-time template parameters so address math folds into
// immediate offsets and the k-loop fully unrolls into a clause-able load
// stream feeding the WMMA chain.
// ---------------------------------------------------------------------------

typedef __attribute__((ext_vector_type(2))) float v2f;
typedef __attribute__((ext_vector_type(8))) float v8f;

// One wave computes a 16x16 tile of D = A[N x K] @ B[K x C] (row-major,
// ldb == ldo == C). blockDim.x = 32 * (C/16): wave w handles column tile w.
// fp32 WMMA operand layout (ISA 7.12.2, 32-bit A 16x4):
//   lanes 0-15:  regs hold K = k0, k0+1  for row M=lane
//   lanes 16-31: regs hold K = k0+2, k0+3 for row M=lane-16
// B (4x16) mirrors this with rows striped across lanes.
template <int K, int C>
__global__ void gemm_wmma_f32(const float* __restrict__ A,
                              const float* __restrict__ B,
                              float* __restrict__ D, int Nrows) {
  const int lane = threadIdx.x & 31;
  const int wv   = threadIdx.x >> 5;
  const int row0 = blockIdx.x * 16;
  const int col0 = wv * 16;
  const int half = lane >> 4;   // 0 for lanes 0-15, 1 for lanes 16-31
  const int l    = lane & 15;

  const int arow = min(row0 + l, Nrows - 1);   // clamp load row (EXEC stays all-1)
  const float* ab = A + (size_t)arow * K + half * 2;   // k-offsets are immediates
  const float* bb = B + col0 + l + half * 2 * C;

  v8f acc = {};
#pragma unroll
  for (int k0 = 0; k0 < K; k0 += 4) {
    const v2f a = *(const v2f*)(ab + k0);              // 8B aligned (k0, half*2 even)
    v2f b;
    b.x = bb[(size_t)k0 * C];                          // B[k0+half*2    ][col]
    b.y = bb[(size_t)(k0 + 1) * C];                    // B[k0+half*2 + 1][col]
    acc = __builtin_amdgcn_wmma_f32_16x16x4_f32(
        /*neg_a=*/false, a, /*neg_b=*/false, b,
        /*c_mod=*/(short)0, acc, /*reuse_a=*/false, /*reuse_b=*/false);
  }
  // D layout: VGPR i -> row M=i (lanes 0-15) / M=8+i (lanes 16-31), col N=l.
  const int srow = row0 + half * 8;
  float* db = D + (size_t)srow * C + col0 + l;
#pragma unroll
  for (int i = 0; i < 8; ++i) {
    if (srow + i < Nrows) db[(size_t)i * C] = acc[i];
  }
}

// Edge scatter: agg[dst[e]] += h[src[e]] * vals[e], float4-vectorized over
// channel groups. GROUPS = #float4 groups per edge, STRIDE4 = row stride in
// float4 units. Native f32 atomics (L2-resident destination).
template <int GROUPS, int STRIDE4>
__global__ void scatter_edges(const float* __restrict__ h,
                              const int* __restrict__ src,
                              const int* __restrict__ dst,
                              const float* __restrict__ vals,
                              float* __restrict__ agg, int E) {
  const int t = blockIdx.x * blockDim.x + threadIdx.x;
  if (t >= E * GROUPS) return;
  const int e = t / GROUPS;          // constant divisor: shift or cheap magic-mul
  const int g = t - e * GROUPS;
  const float v = vals[e];
  const float4 m = *((const float4*)h + (size_t)src[e] * STRIDE4 + g);
  float* ap = agg + ((size_t)dst[e] * STRIDE4 + g) * 4;
  unsafeAtomicAdd(ap + 0, m.x * v);
  unsafeAtomicAdd(ap + 1, m.y * v);
  unsafeAtomicAdd(ap + 2, m.z * v);
  unsafeAtomicAdd(ap + 3, m.w * v);
}

// out = elu(agg + bias); C is a power of two (32 or 64)
template <int C>
__global__ void bias_elu(const float* __restrict__ agg,
                         const float* __restrict__ bias,
                         float* __restrict__ out, int total) {
  const int t = blockIdx.x * blockDim.x + threadIdx.x;
  if (t >= total) return;
  const float z = agg[t] + bias[t & (C - 1)];
  out[t] = z > 0.0f ? z : (expf(z) - 1.0f);
}

// Zero-pad W4 [64x19] -> [64x32]
__global__ void pad_w4(const float* __restrict__ W4, float* __restrict__ Wp) {
  const int t = blockIdx.x * blockDim.x + threadIdx.x;
  if (t >= 64 * 32) return;
  const int k = t >> 5, c = t & 31;
  Wp[t] = (c < 19) ? W4[k * 19 + c] : 0.0f;
}

// Softmax over 19 classes: out[n] = softmax(agg[n, 0:19] + b4), agg stride 32.
__global__ void bias_softmax19(const float* __restrict__ agg,
                               const float* __restrict__ bias,
                               float* __restrict__ out, int N) {
  const int n = blockIdx.x * blockDim.x + threadIdx.x;
  if (n >= N) return;
  const float* r = agg + (size_t)n * 32;
  float v[19];
  float m = -INFINITY;
#pragma unroll
  for (int i = 0; i < 19; ++i) { v[i] = r[i] + bias[i]; m = fmaxf(m, v[i]); }
  float s = 0.0f;
#pragma unroll
  for (int i = 0; i < 19; ++i) { v[i] = expf(v[i] - m); s += v[i]; }
  const float inv = 1.0f / s;
  float* o = out + (size_t)n * 19;
#pragma unroll
  for (int i = 0; i < 19; ++i) o[i] = v[i] * inv;
}

extern "C" void kernel_launch(void* const* d_in, const int* in_sizes, int n_in,
                              void* d_out, int out_size, void* d_ws, size_t ws_size,
                              hipStream_t stream) {
  const float* x     = (const float*)d_in[0];
  const int*   esrc  = (const int*)  d_in[1];
  const int*   edst  = (const int*)  d_in[2];
  const float* evals = (const float*)d_in[3];
  const float* W1    = (const float*)d_in[4];
  const float* b1    = (const float*)d_in[5];
  const float* W2    = (const float*)d_in[6];
  const float* b2    = (const float*)d_in[7];
  const float* W4    = (const float*)d_in[8];
  const float* b4    = (const float*)d_in[9];
  float* out = (float*)d_out;

  const int N = in_sizes[0] / 64;
  const int E = in_sizes[1];

  // Workspace: two N x 64 ping-pong buffers + padded W4 (~51.2 MB total).
  float* bufA = (float*)d_ws;
  float* bufB = bufA + (size_t)N * 64;
  float* w4p  = bufB + (size_t)N * 64;

  const int tiles = (N + 15) / 16;

  // ---- Layer 1: h1 = x @ W1  [N x 32] ----
  gemm_wmma_f32<64, 32><<<tiles, 64, 0, stream>>>(x, W1, bufA, N);
  hipMemsetAsync(bufB, 0, (size_t)N * 32 * sizeof(float), stream);
  {
    const int total = E * 8;
    scatter_edges<8, 8><<<(total + 255) / 256, 256, 0, stream>>>(
        bufA, esrc, edst, evals, bufB, E);
  }
  {
    const int total = N * 32;
    bias_elu<32><<<(total + 255) / 256, 256, 0, stream>>>(bufB, b1, bufA, total);
  }

  // ---- Layer 2: h2 = h1 @ W2  [N x 64] ----
  gemm_wmma_f32<32, 64><<<tiles, 128, 0, stream>>>(bufA, W2, bufB, N);
  hipMemsetAsync(bufA, 0, (size_t)N * 64 * sizeof(float), stream);
  {
    const int total = E * 16;
    scatter_edges<16, 16><<<(total + 255) / 256, 256, 0, stream>>>(
        bufB, esrc, edst, evals, bufA, E);
  }
  {
    const int total = N * 64;
    bias_elu<64><<<(total + 255) / 256, 256, 0, stream>>>(bufA, b2, bufB, total);
  }

  // ---- Layer 3: h3 = h2 @ W4pad  [N x 32], valid channels 0..18 ----
  pad_w4<<<8, 256, 0, stream>>>(W4, w4p);
  gemm_wmma_f32<64, 32><<<tiles, 64, 0, stream>>>(bufB, w4p, bufA, N);
  hipMemsetAsync(bufB, 0, (size_t)N * 32 * sizeof(float), stream);
  {
    // 5 float4 groups -> channels 0..19 (channel 19 is exactly zero from padding)
    const int total = E * 5;
    scatter_edges<5, 8><<<(total + 255) / 256, 256, 0, stream>>>(
        bufA, esrc, edst, evals, bufB, E);
  }
  bias_softmax19<<<(N + 255) / 256, 256, 0, stream>>>(bufB, b4, out, N);
}